// kernel_generated_0_21887153341406
// MI455X (gfx1250) — compile-verified
//
#include <hip/hip_runtime.h>
#include <hip/hip_bf16.h>

// out[i,o,n] = sum_{k,l,j} x[j, n, ll(k,l,o)] * W[i,k,l,j], rolls folded into
// addressing. GEMM view: rows i (64), cols c = o*56+n (3136), reduction
// r = (k*2+l)*64 + j (384). A (64x384) is exactly W flattened.
// fp32 WMMA 16x16x4 keeps reference precision.
//
// K-loop is restructured as 6 (k,l) blocks x 16 WMMA steps: within a block the
// tap row `ll` and its padding predicate are loop-invariant, so B loads become
// immediate-offset global_load_b32 off a single base and padding is a single
// per-block 0/1 float mask (branch-free, no OOB speculation).

typedef __attribute__((ext_vector_type(2))) float v2f;
typedef __attribute__((ext_vector_type(8))) float v8f;

#define CH    64          // output channels (GEMM rows)
#define SP    56          // spatial extent
#define NCOL  (SP * SP)   // 3136 GEMM columns
#define KRED  384         // 3 taps * 2 * 64 input channels
#define PLANE 3136        // elements per input channel plane

__global__ __launch_bounds__(256)
void shiftconv_wmma_f32(const float* __restrict__ x,   // (64,56,56): j*3136 + n*56 + ll
                        const float* __restrict__ W,   // (64,3,2,64) == (64,384) row-major
                        float* __restrict__ out)       // (64,56,56)
{
    const int lane = threadIdx.x & 31;
    const int wave = threadIdx.x >> 5;
    const int tile = blockIdx.x * 8 + wave;      // 784 tiles total
    const int row_tile = tile / 196;             // 0..3   (i tiles of 16)
    const int col_tile = tile % 196;             // 0..195 (c tiles of 16)
    const int row_base = row_tile * 16;
    const int col_base = col_tile * 16;

    const int m     = lane & 15;                 // A row in tile / B col in tile
    const int kh    = lane >> 4;                 // K half selector
    const int kbase = kh * 2;                    // lane K pair: {kbase, kbase+1}

    // Per-lane B column geometry (fixed for the whole kernel).
    const int c    = col_base + m;
    const int o    = c / 56;
    const int n    = c % 56;
    const int xrow = n * 56;                     // row-n offset inside a channel plane
    const int ll0  = o - 1;                      // l=0 tap base (+k)
    const int ll1  = ((o + 55) % 56) - 1;        // l=1 tap base (+k), roll folded in

    const float* Arow = W + (row_base + m) * KRED + kbase;

    v8f acc = {};                                // C = 0

    #pragma unroll
    for (int kl = 0; kl < 6; ++kl) {
        const int k = kl >> 1;
        const int l = kl & 1;
        const int ll = (l ? ll1 : ll0) + k;

        // Branch-free zero padding: clamp address into range, scale by 0/1.
        const bool  valid = (unsigned)ll < 56u;
        const int   llc   = valid ? ll : 0;
        const float fmask = valid ? 1.0f : 0.0f;

        // Lane's two B rows in this block start at j = kbase, kbase+1.
        const float* bp0 = x + (size_t)(kbase + 0) * PLANE + xrow + llc;
        const float* bp1 = x + (size_t)(kbase + 1) * PLANE + xrow + llc;
        const float* ap  = Arow + kl * 64;

        #pragma unroll
        for (int q0 = 0; q0 < 64; q0 += 4) {     // 16 WMMA steps, j advances by 4
            v2f a = *(const v2f*)(ap + q0);      // global_load_b64, imm offset

            v2f b;
            b[0] = bp0[q0 * PLANE] * fmask;      // global_load_b32, imm offset
            b[1] = bp1[q0 * PLANE] * fmask;

            acc = __builtin_amdgcn_wmma_f32_16x16x4_f32(
                /*neg_a=*/false, a, /*neg_b=*/false, b,
                /*c_mod=*/(short)0, acc, /*reuse_a=*/false, /*reuse_b=*/false);
        }
    }

    // D layout: VGPR vr <-> row (vr + 8*kh), col m. Fold final roll(+1) on n.
    const int ostore = o * 56 + ((n + 1) % 56);
    #pragma unroll
    for (int vr = 0; vr < 8; ++vr) {
        const int i = row_base + vr + 8 * kh;
        out[i * NCOL + ostore] = acc[vr];
    }
}

extern "C" void kernel_launch(void* const* d_in, const int* in_sizes, int n_in,
                              void* d_out, int out_size, void* d_ws, size_t ws_size,
                              hipStream_t stream) {
    const float* x = (const float*)d_in[0];   // 1*64*56*56 = 200704 fp32
    const float* W = (const float*)d_in[1];   // 64*3*2*64  = 24576  fp32
    float* out = (float*)d_out;               // 200704 fp32

    // 784 tiles of 16x16, 8 waves (tiles) per 256-thread block -> 98 blocks.
    shiftconv_wmma_f32<<<98, 256, 0, stream>>>(x, W, out);
}